// GutzwillerWaveFunction_36077725287033
// MI455X (gfx1250) — compile-verified
//
#include <hip/hip_runtime.h>
#include <math.h>

#define BATCH 512
#define NM    512
#define NE    256
#define NB    32

typedef float v8f __attribute__((ext_vector_type(8)));
typedef float v2f __attribute__((ext_vector_type(2)));

static __device__ inline v8f wmma4(v2f a, v2f b, v8f c) {
  // D = A(16x4) * B(4x16) + C(16x16), fp32
  return __builtin_amdgcn_wmma_f32_16x16x4_f32(false, a, false, b, (short)0, c, false, false);
}

// ---------------------------------------------------------------------------
// Kernel 1: per-batch gather of occupied orbitals into planar fp32 re/im
// planes, plus double-occupancy flag.  One workgroup (256 threads) per batch.
// ---------------------------------------------------------------------------
__global__ __launch_bounds__(256) void gutz_gather(const int* __restrict__ x,
                                                   const float* __restrict__ mf,
                                                   float* __restrict__ ws,
                                                   int* __restrict__ doflags) {
  const int b = blockIdx.x;
  const int t = threadIdx.x;

  __shared__ int sx[NM];
  __shared__ int sc[NM];
  __shared__ int occ[NE];
  __shared__ int sdo;

  const int* xr = x + (size_t)b * NM;
  sx[t]       = xr[t];
  sx[t + 256] = xr[t + 256];
  if (t == 0) sdo = 0;
  __syncthreads();

  // double occupancy: mode i and i+256 both set
  if (sx[t] + sx[t + 256] > 1) atomicOr(&sdo, 1);

  // inclusive prefix scan over 512 entries (Hillis-Steele, 2 elems/thread)
  sc[t] = sx[t];
  sc[t + 256] = sx[t + 256];
  __syncthreads();
  for (int off = 1; off < NM; off <<= 1) {
    int v0 = sc[t]       + ((t >= off) ? sc[t - off] : 0);
    int v1 = sc[t + 256] + sc[t + 256 - off];  // t+256 >= off always (off<=256)
    __syncthreads();
    sc[t] = v0;
    sc[t + 256] = v1;
    __syncthreads();
  }
  if (sx[t])       occ[sc[t] - 1]       = t;
  if (sx[t + 256]) occ[sc[t + 256] - 1] = t + 256;
  __syncthreads();

  float* Re = ws + (size_t)b * 2 * NE * NE;
  float* Im = Re + (size_t)NE * NE;
  // copy 256 selected rows of mf (interleaved complex) into planar planes
  for (int e = 0; e < NE; ++e) {
    const float* src = mf + (size_t)occ[e] * NE * 2;
    Re[(size_t)e * NE + t] = src[2 * t];
    Im[(size_t)e * NE + t] = src[2 * t + 1];
  }
  if (t == 0) doflags[b] = sdo;
}

// ---------------------------------------------------------------------------
// Kernel 2: per-batch blocked complex LU with partial pivoting; trailing
// Schur-complement update via V_WMMA_F32_16X16X4_F32 (4 real WMMAs / chunk).
// One workgroup (256 threads = 8 wave32) per batch.
// ---------------------------------------------------------------------------
__global__ __launch_bounds__(256) void gutz_lu(float* __restrict__ ws,
                                               const int* __restrict__ doflags,
                                               float* __restrict__ out) {
  const int b    = blockIdx.x;
  const int t    = threadIdx.x;
  const int lane = t & 31;
  const int w    = t >> 5;       // wave id 0..7
  const int hl   = lane >> 4;    // half-wave select
  const int l15  = lane & 15;

  float* Re = ws + (size_t)b * 2 * NE * NE;
  float* Im = Re + (size_t)NE * NE;

  __shared__ float L11re[NB][NB + 1];
  __shared__ float L11im[NB][NB + 1];
  __shared__ float redV[256];
  __shared__ int   redI[256];
  __shared__ int   pivRow;
  __shared__ float accLog, accPh;

  if (t == 0) { accLog = 0.0f; accPh = 0.0f; }
  __syncthreads();

  for (int k0 = 0; k0 < NE; k0 += NB) {
    // ---------------- panel factorization (columns k0..k0+31) --------------
    for (int j = k0; j < k0 + NB; ++j) {
      // pivot search over rows j..255 (thread t <-> row t)
      float mag = -1.0f;
      if (t >= j) {
        float ar = Re[(size_t)t * NE + j], ai = Im[(size_t)t * NE + j];
        mag = ar * ar + ai * ai;
      }
      redV[t] = mag; redI[t] = t;
      __syncthreads();
      for (int s = 128; s > 0; s >>= 1) {
        if (t < s && redV[t + s] > redV[t]) { redV[t] = redV[t + s]; redI[t] = redI[t + s]; }
        __syncthreads();
      }
      if (t == 0) {
        int p = redI[0];
        pivRow = p;
        float pr = Re[(size_t)p * NE + j], pi = Im[(size_t)p * NE + j];
        accLog += 0.5f * logf(redV[0]);                 // log|U_jj|
        accPh  += atan2f(pi, pr) + ((p != j) ? 3.14159265358979f : 0.0f);
      }
      __syncthreads();
      const int p = pivRow;
      // full-row swap (thread t <-> column t)
      if (p != j) {
        float tr = Re[(size_t)j * NE + t], ti = Im[(size_t)j * NE + t];
        Re[(size_t)j * NE + t] = Re[(size_t)p * NE + t];
        Im[(size_t)j * NE + t] = Im[(size_t)p * NE + t];
        Re[(size_t)p * NE + t] = tr;
        Im[(size_t)p * NE + t] = ti;
      }
      __syncthreads();
      // fused scale (complex divide by pivot) + rank-1 update of panel row t
      if (t > j) {
        float pr  = Re[(size_t)j * NE + j], pi = Im[(size_t)j * NE + j];
        float inv = 1.0f / (pr * pr + pi * pi);
        float ar  = Re[(size_t)t * NE + j], ai = Im[(size_t)t * NE + j];
        float mr  = (ar * pr + ai * pi) * inv;
        float mi  = (ai * pr - ar * pi) * inv;
        Re[(size_t)t * NE + j] = mr;
        Im[(size_t)t * NE + j] = mi;
        for (int jj = j + 1; jj < k0 + NB; ++jj) {
          float ur = Re[(size_t)j * NE + jj], ui = Im[(size_t)j * NE + jj];
          Re[(size_t)t * NE + jj] -= mr * ur - mi * ui;
          Im[(size_t)t * NE + jj] -= mr * ui + mi * ur;
        }
      }
      __syncthreads();
    }

    // ---------------- cache unit-lower L11 block in LDS --------------------
    for (int idx = t; idx < NB * NB; idx += 256) {
      int r = idx >> 5, c = idx & 31;
      L11re[r][c] = Re[(size_t)(k0 + r) * NE + k0 + c];
      L11im[r][c] = Im[(size_t)(k0 + r) * NE + k0 + c];
    }
    __syncthreads();

    const int base = k0 + NB;
    const int MN   = NE - base;

    // ---------------- TRSM: U12 = L11^{-1} * A12 (one column per thread) ---
    if (t < MN) {
      const int c = base + t;
      for (int r = 1; r < NB; ++r) {
        float sr = Re[(size_t)(k0 + r) * NE + c], si = Im[(size_t)(k0 + r) * NE + c];
        for (int m = 0; m < r; ++m) {
          float lr = L11re[r][m], li = L11im[r][m];
          float ar = Re[(size_t)(k0 + m) * NE + c], ai = Im[(size_t)(k0 + m) * NE + c];
          sr -= lr * ar - li * ai;
          si -= lr * ai + li * ar;
        }
        Re[(size_t)(k0 + r) * NE + c] = sr;
        Im[(size_t)(k0 + r) * NE + c] = si;
      }
    }
    __syncthreads();

    // ---------------- trailing update A22 -= L21 * U12 via WMMA ------------
    const int tpr    = MN >> 4;       // 16x16 tiles per row/col
    const int ntiles = tpr * tpr;
    for (int tt = w; tt < ntiles; tt += 8) {   // wave-uniform: EXEC stays full
      const int ti = tt / tpr, tj = tt - ti * tpr;
      const int row0 = base + ti * 16;
      const int col0 = base + tj * 16;
      const int crow = row0 + 8 * hl;          // C: VGPR r -> M = r + 8*half
      const int ccol = col0 + l15;             // C: lane   -> N
      v8f cre, cim;
#pragma unroll
      for (int r = 0; r < 8; ++r) {
        cre[r] = Re[(size_t)(crow + r) * NE + ccol];
        cim[r] = Im[(size_t)(crow + r) * NE + ccol];
      }
      const int arow = row0 + l15;             // A: lane -> M
      for (int kk = 0; kk < NB; kk += 4) {
        const int ka = k0 + kk + 2 * hl;       // A: K = 2*half + v ; B: K = v + 2*half
        v2f are, aim, bre, bim;
        are[0] = Re[(size_t)arow * NE + ka];     are[1] = Re[(size_t)arow * NE + ka + 1];
        aim[0] = Im[(size_t)arow * NE + ka];     aim[1] = Im[(size_t)arow * NE + ka + 1];
        bre[0] = Re[(size_t)ka * NE + ccol];     bre[1] = Re[(size_t)(ka + 1) * NE + ccol];
        bim[0] = Im[(size_t)ka * NE + ccol];     bim[1] = Im[(size_t)(ka + 1) * NE + ccol];
        v2f nare = -are, naim = -aim;
        // Cre += (-Are)*Bre + Aim*Bim ; Cim += (-Are)*Bim + (-Aim)*Bre
        cre = wmma4(nare, bre, cre);
        cre = wmma4(aim,  bim, cre);
        cim = wmma4(nare, bim, cim);
        cim = wmma4(naim, bre, cim);
      }
#pragma unroll
      for (int r = 0; r < 8; ++r) {
        Re[(size_t)(crow + r) * NE + ccol] = cre[r];
        Im[(size_t)(crow + r) * NE + ccol] = cim[r];
      }
    }
    __syncthreads();
  }

  if (t == 0) {
    float ph = atan2f(sinf(accPh), cosf(accPh));   // wrap to (-pi, pi]
    float re = accLog - (doflags[b] ? 1.0e14f : 0.0f);
    out[2 * b]     = re;   // complex64: real
    out[2 * b + 1] = ph;   // complex64: imag
  }
}

// ---------------------------------------------------------------------------
extern "C" void kernel_launch(void* const* d_in, const int* in_sizes, int n_in,
                              void* d_out, int out_size, void* d_ws, size_t ws_size,
                              hipStream_t stream) {
  (void)in_sizes; (void)n_in; (void)out_size; (void)ws_size;
  const int*   x  = (const int*)d_in[0];     // (512, 512) int32 occupancy
  const float* mf = (const float*)d_in[1];   // (512, 256) complex64, interleaved
  float* wsf = (float*)d_ws;                 // 512 * (2 * 256*256) fp32 planes
  int*   doflags = (int*)(wsf + (size_t)BATCH * 2 * NE * NE);
  float* out = (float*)d_out;                // 512 complex64 -> 1024 fp32

  gutz_gather<<<BATCH, 256, 0, stream>>>(x, mf, wsf, doflags);
  gutz_lu<<<BATCH, 256, 0, stream>>>(wsf, doflags, out);
}